// EPLoss_87531433492691
// MI455X (gfx1250) — compile-verified
//
#include <hip/hip_runtime.h>

typedef __attribute__((ext_vector_type(2))) float v2f;
typedef __attribute__((ext_vector_type(8))) float v8f;

#define NY 512   // n_y
#define NT 128   // n_T
#define NC 128   // C
#define EOS_ID 1

// Block-wide inclusive scan of the linear recurrence x_j = A_j + M_j * x_{j-1}
// over 512 threads (16 waves of 32). Element 0 must carry M=0 so x_{-1} dies.
// Returns x_tid. Contains 2 __syncthreads.
__device__ __forceinline__ float block_linrec_scan(float M, float A,
                                                   float* swm, float* swa,
                                                   float* swaI) {
  const int tid  = threadIdx.x;
  const int lane = tid & 31;
  const int wv   = tid >> 5;
  // intra-wave segmented scan via shuffles (wave32)
  #pragma unroll
  for (int d = 1; d < 32; d <<= 1) {
    float Mu = __shfl_up(M, d, 32);
    float Au = __shfl_up(A, d, 32);
    if (lane >= d) { A = fmaf(M, Au, A); M = M * Mu; }
  }
  if (lane == 31) { swm[wv] = M; swa[wv] = A; }
  __syncthreads();
  // wave 0 scans the 16 per-wave segment operators
  if (tid < 16) {
    float m2 = swm[tid], a2 = swa[tid];
    #pragma unroll
    for (int d = 1; d < 16; d <<= 1) {
      float Mu = __shfl_up(m2, d, 32);
      float Au = __shfl_up(a2, d, 32);
      if (tid >= d) { a2 = fmaf(m2, Au, a2); m2 = m2 * Mu; }
    }
    swaI[tid] = a2;  // inclusive A for waves 0..15 (M collapses: element 0 has M=0)
  }
  __syncthreads();
  return (wv == 0) ? A : fmaf(M, swaI[wv - 1], A);
}

__global__ __launch_bounds__(512) void ep_dp_kernel(
    const float* __restrict__ pred,  // [B, NY, NC]
    const float* __restrict__ Rg,    // [B, NY, 3]
    const float* __restrict__ Ins,   // [B, NY, NC]
    const int*   __restrict__ tgt,   // [B, NT]
    float* __restrict__ out)         // [B]
{
  __shared__ float gp[16][NY];   // g_pred rows i0..i0+15, 32 KB
  __shared__ float gi[16][NY];   // g_ins  rows i0..i0+15, 32 KB
  __shared__ float rcopy[NY], rins[NY], rdel[NY];
  __shared__ float ep[NY];
  __shared__ int   tg[NT];
  __shared__ float swm[16], swa[16], swaI[16];

  const int b    = blockIdx.x;
  const int tid  = threadIdx.x;
  const int lane = tid & 31;
  const int wv   = tid >> 5;

  // --- stage R (with R_ins[last] := 1) and target ---
  const float* Rb = Rg + (size_t)b * NY * 3;
  rcopy[tid] = Rb[tid * 3 + 0];
  rins[tid]  = (tid == NY - 1) ? 1.0f : Rb[tid * 3 + 1];
  rdel[tid]  = Rb[tid * 3 + 2];
  if (tid < NT) tg[tid] = tgt[b * NT + tid];
  __syncthreads();

  // --- row 0: ep[0][j] = prod_{k=1..j} p_D[0,k] ---
  {
    const bool eos0 = (tg[0] == EOS_ID);
    float M = (tid == 0) ? 0.0f : (eos0 ? 1.0f : rdel[tid]);
    float A = (tid == 0) ? 1.0f : 0.0f;
    float x = block_linrec_scan(M, A, swm, swa, swaI);
    ep[tid] = x;
    __syncthreads();
  }

  const float* predb = pred + (size_t)b * NY * NC;
  const float* insb  = Ins  + (size_t)b * NY * NC;
  const int halfh = lane >> 4;           // K-half for A/B fragments

  for (int ci = 0; ci < NT / 16; ++ci) {
    const int i0 = ci * 16;
    // ---- G chunk via one-hot f32 WMMA: G[m,j] = sum_c 1[c==tg[m]] * src[j,c] ----
    const int tcol = tg[i0 + (lane & 15)];   // A fragment: M = lane%16
    #pragma unroll
    for (int tsel = 0; tsel < 2; ++tsel) {
      const int jt = wv * 2 + tsel;          // 16 waves x 2 tiles = 32 tiles = NY
      const int jn = jt * 16 + (lane & 15);  // B fragment: N = lane%16
      const float* bp0 = predb + (size_t)jn * NC + (halfh << 1);
      const float* bi0 = insb  + (size_t)jn * NC + (halfh << 1);
      v8f accp = {};
      v8f acci = {};
      // Cap unroll: full unroll preloaded ~128 VGPRs of B fragments and
      // spilled to scratch mid-WMMA-stream. 4x keeps loads pipelined with
      // no spills.
      #pragma unroll 4
      for (int k = 0; k < NC; k += 4) {
        const int kk = k + (halfh << 1);
        v2f a;
        a.x = (tcol == kk)     ? 1.0f : 0.0f;
        a.y = (tcol == kk + 1) ? 1.0f : 0.0f;
        v2f bp = *(const v2f*)(bp0 + k);
        v2f bi = *(const v2f*)(bi0 + k);
        accp = __builtin_amdgcn_wmma_f32_16x16x4_f32(false, a, false, bp,
                                                     (short)0, accp, false, false);
        acci = __builtin_amdgcn_wmma_f32_16x16x4_f32(false, a, false, bi,
                                                     (short)0, acci, false, false);
      }
      // D layout: VGPR v, half h -> M = v + 8h ; N = lane%16
      #pragma unroll
      for (int v = 0; v < 8; ++v) {
        gp[v + halfh * 8][jn] = accp[v];
        gi[v + halfh * 8][jn] = acci[v];
      }
    }
    __syncthreads();

    // ---- DP rows i = i0+1 .. i0+16 consume G rows i-1 = i0+r ----
    for (int r = 0; r < 16; ++r) {
      const int i = i0 + r + 1;
      if (i >= NT) break;                    // uniform branch
      const bool eos = (tg[i] == EOS_ID);
      const float epj = ep[tid];
      float M, A;
      if (tid == 0) {
        // ep[i,0] = ep[i-1,0] * p_I[i-1,1]
        A = epj * rins[1] * gi[r][1];
        M = 0.0f;
      } else {
        const float epjm1 = ep[tid - 1];
        const float pc = rcopy[tid - 1] * gp[r][tid - 1];  // p_C[i-1, j-1]
        const float pi = rins[tid] * gi[r][tid];           // p_I[i-1, j]
        A = epjm1 * pc + epj * pi;
        M = eos ? 1.0f : rdel[tid - 1];                    // p_D[i, j-1]
      }
      const float x = block_linrec_scan(M, A, swm, swa, swaI);
      ep[tid] = x;
      __syncthreads();
    }
    __syncthreads();  // gp/gi rewritten next chunk
  }

  if (tid == 0) out[b] = ep[NY - 1];
}

extern "C" void kernel_launch(void* const* d_in, const int* in_sizes, int n_in,
                              void* d_out, int out_size, void* d_ws, size_t ws_size,
                              hipStream_t stream) {
  (void)n_in; (void)d_ws; (void)ws_size; (void)out_size;
  const float* pred = (const float*)d_in[0];
  const float* Rg   = (const float*)d_in[1];
  const float* Ins  = (const float*)d_in[2];
  const int*   tgt  = (const int*)d_in[3];
  float* out = (float*)d_out;
  const int B = in_sizes[3] / NT;   // target is [B, NT]
  ep_dp_kernel<<<B, NY, 0, stream>>>(pred, Rg, Ins, tgt, out);
}